// QSendLayerGPU_21388937134848
// MI455X (gfx1250) — compile-verified
//
#include <hip/hip_runtime.h>
#include <stdint.h>

typedef float v4f __attribute__((ext_vector_type(4)));

#define QTILE 2048   // floats per block per pipeline stage (256 thr x 2 x float4)

// ---------------------------------------------------------------------------
// Pass 1: per-block min/max partials (grid-stride, b128 loads, wave32 reduce)
// ---------------------------------------------------------------------------
__global__ __launch_bounds__(256) void k_minmax(const v4f* __restrict__ x4,
                                                long long n4,
                                                float* __restrict__ partials)
{
    float vmin = __builtin_inff();
    float vmax = -__builtin_inff();
    const long long stride = (long long)gridDim.x * blockDim.x;
    for (long long i = (long long)blockIdx.x * blockDim.x + threadIdx.x; i < n4; i += stride) {
        v4f v = x4[i];
        vmin = fminf(vmin, fminf(fminf(v.x, v.y), fminf(v.z, v.w)));
        vmax = fmaxf(vmax, fmaxf(fmaxf(v.x, v.y), fmaxf(v.z, v.w)));
    }
    // wave32 reduction
    #pragma unroll
    for (int off = 16; off > 0; off >>= 1) {
        vmin = fminf(vmin, __shfl_xor(vmin, off, 32));
        vmax = fmaxf(vmax, __shfl_xor(vmax, off, 32));
    }
    __shared__ float smin[8], smax[8];
    const int wave = threadIdx.x >> 5;
    if ((threadIdx.x & 31) == 0) { smin[wave] = vmin; smax[wave] = vmax; }
    __syncthreads();
    if (threadIdx.x == 0) {
        float m = smin[0], M = smax[0];
        #pragma unroll
        for (int w = 1; w < 8; ++w) { m = fminf(m, smin[w]); M = fmaxf(M, smax[w]); }
        partials[2 * blockIdx.x]     = m;
        partials[2 * blockIdx.x + 1] = M;
    }
}

// ---------------------------------------------------------------------------
// Pass 2: reduce block partials, compute mn/step, publish scalars
// ---------------------------------------------------------------------------
__global__ __launch_bounds__(256) void k_finalize(const float* __restrict__ partials, int nb,
                                                  float* __restrict__ scal,
                                                  float* __restrict__ tail)
{
    float m = __builtin_inff();
    float M = -__builtin_inff();
    for (int i = threadIdx.x; i < nb; i += 256) {
        m = fminf(m, partials[2 * i]);
        M = fmaxf(M, partials[2 * i + 1]);
    }
    #pragma unroll
    for (int off = 16; off > 0; off >>= 1) {
        m = fminf(m, __shfl_xor(m, off, 32));
        M = fmaxf(M, __shfl_xor(M, off, 32));
    }
    __shared__ float smin[8], smax[8];
    const int wave = threadIdx.x >> 5;
    if ((threadIdx.x & 31) == 0) { smin[wave] = m; smax[wave] = M; }
    __syncthreads();
    if (threadIdx.x == 0) {
        float mm = smin[0], MM = smax[0];
        #pragma unroll
        for (int w = 1; w < 8; ++w) { mm = fminf(mm, smin[w]); MM = fmaxf(MM, smax[w]); }
        const float step = (MM - mm) * (1.0f / 255.0f);
        scal[0] = mm;
        scal[1] = step;
        scal[2] = 1.0f / step;
        tail[0] = mm;     // d_out[2N]   = mn
        tail[1] = step;   // d_out[2N+1] = step
    }
}

// ---------------------------------------------------------------------------
// Pass 3: quantize + dequantize.
// Triple-buffered async global->LDS pipeline: two tiles (4 async b128 loads)
// in flight per wave; partial wait s_wait_asynccnt 2 retires only the oldest
// tile (async loads complete in order), keeping the load pipe busy across the
// wait/barrier boundary. Non-temporal b128 stores keep x resident in L2.
// ---------------------------------------------------------------------------
__global__ __launch_bounds__(256) void k_quant(const float* __restrict__ x,
                                               float* __restrict__ dq,
                                               float* __restrict__ qf,
                                               const float* __restrict__ scal,
                                               long long nTiles)
{
    __shared__ float smem[3][QTILE];
    const float mn       = scal[0];
    const float step     = scal[1];
    const float inv_step = scal[2];
    const int tid = threadIdx.x;

    auto issue = [&](long long t, int b) {
        const float* g0 = x + t * QTILE + (tid << 2);
        const float* g1 = g0 + 1024;
        // Low 32 bits of a generic LDS pointer == LDS byte offset (flat aperture rule)
        unsigned l0 = (unsigned)(uintptr_t)&smem[b][tid << 2];
        unsigned l1 = l0 + 1024u * 4u;
        asm volatile("global_load_async_to_lds_b128 %0, %1, off"
                     :: "v"(l0), "v"(g0) : "memory");
        asm volatile("global_load_async_to_lds_b128 %0, %1, off"
                     :: "v"(l1), "v"(g1) : "memory");
    };

    long long t = blockIdx.x;
    const long long ts = gridDim.x;
    int buf = 0;

    if (t < nTiles)      issue(t, 0);
    if (t + ts < nTiles) issue(t + ts, 1);

    while (t < nTiles) {
        // Retire only the oldest tile's loads when a successor is in flight
        // (uniform scalar branch: t, ts, nTiles are wave-uniform).
        if (t + ts < nTiles) {
            asm volatile("s_wait_asynccnt 2" ::: "memory");
        } else {
            asm volatile("s_wait_asynccnt 0" ::: "memory");
        }
        __syncthreads();

        const long long t2 = t + 2 * ts;
        if (t2 < nTiles) issue(t2, buf + 2 >= 3 ? buf - 1 : buf + 2);

        const long long base = t * QTILE;
        #pragma unroll
        for (int h = 0; h < 2; ++h) {
            const int lidx = h * 1024 + (tid << 2);
            v4f v = *(const v4f*)&smem[buf][lidx];

            v4f oq, odq;
            {
                float q0 = rintf((v.x - mn) * inv_step) - 128.0f;
                float q1 = rintf((v.y - mn) * inv_step) - 128.0f;
                float q2 = rintf((v.z - mn) * inv_step) - 128.0f;
                float q3 = rintf((v.w - mn) * inv_step) - 128.0f;
                q0 = fminf(fmaxf(q0, -128.0f), 127.0f);
                q1 = fminf(fmaxf(q1, -128.0f), 127.0f);
                q2 = fminf(fmaxf(q2, -128.0f), 127.0f);
                q3 = fminf(fmaxf(q3, -128.0f), 127.0f);
                oq.x = q0; oq.y = q1; oq.z = q2; oq.w = q3;
                odq.x = fmaf(q0 + 128.0f, step, mn);
                odq.y = fmaf(q1 + 128.0f, step, mn);
                odq.z = fmaf(q2 + 128.0f, step, mn);
                odq.w = fmaf(q3 + 128.0f, step, mn);
            }
            const long long gi = base + lidx;
            __builtin_nontemporal_store(odq, (v4f*)(dq + gi));   // th:TH_STORE_NT
            __builtin_nontemporal_store(oq,  (v4f*)(qf + gi));
        }
        t += ts;
        buf = (buf + 1 >= 3) ? 0 : buf + 1;
    }
}

// scalar tail (n not a multiple of QTILE; unused for the 64Mi reference shape)
__global__ void k_quant_tail(const float* __restrict__ x,
                             float* __restrict__ dq, float* __restrict__ qf,
                             const float* __restrict__ scal,
                             long long start, long long n)
{
    long long i = start + (long long)blockIdx.x * blockDim.x + threadIdx.x;
    if (i >= n) return;
    const float mn = scal[0], step = scal[1], inv_step = scal[2];
    float q = rintf((x[i] - mn) * inv_step) - 128.0f;
    q = fminf(fmaxf(q, -128.0f), 127.0f);
    dq[i] = fmaf(q + 128.0f, step, mn);
    qf[i] = q;
}

extern "C" void kernel_launch(void* const* d_in, const int* in_sizes, int n_in,
                              void* d_out, int out_size, void* d_ws, size_t ws_size,
                              hipStream_t stream)
{
    const float* x = (const float*)d_in[0];
    const long long n = (long long)in_sizes[0];

    float* out  = (float*)d_out;
    float* dq   = out;            // [0, n)
    float* qf   = out + n;        // [n, 2n)
    float* tail = out + 2 * n;    // mn, step

    float* ws       = (float*)d_ws;
    float* scal     = ws;         // mn, step, 1/step
    float* partials = ws + 16;    // 2 * NB1 floats

    const int NB1 = 1024;
    k_minmax<<<NB1, 256, 0, stream>>>((const v4f*)x, n >> 2, partials);
    k_finalize<<<1, 256, 0, stream>>>(partials, NB1, scal, tail);

    const long long nTiles = n / QTILE;
    if (nTiles > 0) {
        int NB3 = 4096;
        if (nTiles < NB3) NB3 = (int)nTiles;
        k_quant<<<NB3, 256, 0, stream>>>(x, dq, qf, scal, nTiles);
    }
    const long long start = nTiles * QTILE;
    const long long rem = n - start;
    if (rem > 0) {
        int blocks = (int)((rem + 255) / 256);
        k_quant_tail<<<blocks, 256, 0, stream>>>(x, dq, qf, scal, start, n);
    }
}